// TemporalAttention_32624571580870
// MI455X (gfx1250) — compile-verified
//
#include <hip/hip_runtime.h>
#include <hip/hip_bf16.h>

// TemporalAttention for MI455X (gfx1250, wave32).
// Memory-bound: ~503MB HBM traffic -> ~22us floor @ 23.3 TB/s.
// GEMMs use V_WMMA_F32_16X16X4_F32 (full fp32 precision, CDNA5 matrix path).
// Streaming kernels use float4 (b128) accesses to minimize vmem issue count.

typedef __attribute__((ext_vector_type(2))) float v2f;
typedef __attribute__((ext_vector_type(8))) float v8f;

#define BT   40      // B*T
#define C_   256
#define NPOS 640     // BT*16 spatial positions
#define NH_  4
#define HD_  64
#define T__  5

// ---------------------------------------------------------------------------
// Kernel 1: adaptive avg pool 64x64 -> 4x4 (exact 16x16 block means).
// One block per (bt,c) plane. Each thread sums one 16-float (64B-aligned)
// row segment via 4x b128 loads. Output layout [c][n], n = bt*16 + (by*4+bx),
// i.e. directly the B-matrix layout for the QKV GEMM.
// ---------------------------------------------------------------------------
__global__ __launch_bounds__(256) void pool_kernel(const float* __restrict__ x,
                                                   float* __restrict__ xp) {
  int plane = blockIdx.x;            // bt*256 + c
  int bt = plane >> 8, c = plane & 255;
  int tid = threadIdx.x;
  int cell = tid >> 4;               // 0..15 output cell
  int j    = tid & 15;               // row-within-block
  int by = cell >> 2, bx = cell & 3;
  const float4* p4 = (const float4*)(x + (size_t)plane * 4096 +
                                     (by * 16 + j) * 64 + bx * 16);
  float s = 0.f;
#pragma unroll
  for (int i = 0; i < 4; ++i) {
    float4 v = p4[i];
    s += (v.x + v.y) + (v.z + v.w);
  }
  // reduce the 16 row-sums of this cell (16-lane subgroups of the wave32)
#pragma unroll
  for (int off = 1; off < 16; off <<= 1) s += __shfl_xor(s, off, 16);
  if (j == 0) xp[c * NPOS + bt * 16 + cell] = s * (1.f / 256.f);
}

// ---------------------------------------------------------------------------
// Kernel 2: LayerNorm over C per position (biased var, eps=1e-6).
// One block (256 threads = 256 channels) per position n. Data is L2-resident.
// ---------------------------------------------------------------------------
__global__ __launch_bounds__(256) void ln_kernel(const float* __restrict__ xp,
                                                 const float* __restrict__ gamma,
                                                 const float* __restrict__ beta,
                                                 float* __restrict__ xn) {
  int n = blockIdx.x;                // 0..639
  int c = threadIdx.x;
  float v = xp[c * NPOS + n];
  __shared__ float s1[256], s2[256];
  s1[c] = v; s2[c] = v * v;
  __syncthreads();
  for (int off = 128; off > 0; off >>= 1) {
    if (c < off) { s1[c] += s1[c + off]; s2[c] += s2[c + off]; }
    __syncthreads();
  }
  float mean = s1[0] * (1.f / 256.f);
  float var  = s2[0] * (1.f / 256.f) - mean * mean;
  float rstd = rsqrtf(var + 1e-6f);
  xn[c * NPOS + n] = (v - mean) * rstd * gamma[c] + beta[c];
}

// ---------------------------------------------------------------------------
// Kernel 3/5: fp32 WMMA GEMM.  D[M][N] = A[M][K] @ B[K][N] + bias[M].
// One wave per 16x16 tile, K-loop of V_WMMA_F32_16X16X4_F32.
// A 16x4 layout: lane L holds row M=L&15; VGPR0/1 = K = 2*(L>>4) + {0,1}.
// B 4x16 layout: lane L holds col N=L&15; VGPR0/1 = K = 2*(L>>4) + {0,1}.
// D: VGPR r = element (M = r + 8*(L>>4), N = L&15).
// ---------------------------------------------------------------------------
__global__ __launch_bounds__(32) void gemm_f32_wmma(const float* __restrict__ A,
                                                    const float* __restrict__ B,
                                                    const float* __restrict__ bias,
                                                    float* __restrict__ D,
                                                    int M, int N, int K) {
  int ntiles = N >> 4;
  int mt = blockIdx.x / ntiles;
  int nt = blockIdx.x % ntiles;
  int lane = threadIdx.x & 31;
  int half = lane >> 4;              // which K-pair
  int lm   = lane & 15;              // M for A-frag, N for B-frag
  const float* Arow = A + (size_t)(mt * 16 + lm) * K;
  const float* Bcol = B + (nt * 16 + lm);
  v8f acc = {};
  for (int k = 0; k < K; k += 4) {
    int ka = k + 2 * half;
    v2f a, b;
    a.x = Arow[ka];
    a.y = Arow[ka + 1];
    b.x = Bcol[(size_t)ka * N];
    b.y = Bcol[(size_t)(ka + 1) * N];
    acc = __builtin_amdgcn_wmma_f32_16x16x4_f32(
        /*neg_a=*/false, a, /*neg_b=*/false, b,
        /*c_mod=*/(short)0, acc, /*reuse_a=*/false, /*reuse_b=*/false);
  }
#pragma unroll
  for (int r = 0; r < 8; ++r) {
    int m = mt * 16 + r + 8 * half;
    D[(size_t)m * N + nt * 16 + lm] = acc[r] + bias[m];
  }
}

// ---------------------------------------------------------------------------
// Kernel 4: attention per (b,h). q,k,v are [80][64] tiles (80 = T*16).
// Temporal pos bias p[h,d,t] = pos[h*320 + d*5 + t] added to q and k.
// ~80 MFLOP total across all 32 heads -> scalar VALU in LDS is fine.
// ---------------------------------------------------------------------------
__global__ __launch_bounds__(128) void attn_kernel(const float* __restrict__ qkv, // [768][640]
                                                   const float* __restrict__ pos, // [T*C]
                                                   float* __restrict__ y) {       // [256][640]
  int b = blockIdx.x >> 2;
  int h = blockIdx.x & 3;
  __shared__ float qs[80][64], ks[80][64], vs[80][64];   // 60 KB LDS
  int tid = threadIdx.x;
  for (int idx = tid; idx < 80 * 64; idx += 128) {
    int r = idx >> 6;                 // 0..79 = t*16 + s
    int d = idx & 63;
    int t = r >> 4, sp = r & 15;
    int n = (b * T__ + t) * 16 + sp;
    float p = pos[h * (HD_ * T__) + d * T__ + t];
    int o = h * HD_ + d;
    qs[r][d] = qkv[(size_t)(o)        * NPOS + n] + p;
    ks[r][d] = qkv[(size_t)(o + 256)  * NPOS + n] + p;
    vs[r][d] = qkv[(size_t)(o + 512)  * NPOS + n];
  }
  __syncthreads();
  if (tid < 80) {
    int r = tid;
    // pass 1: row max of scaled scores
    float mx = -1e30f;
    for (int c2 = 0; c2 < 80; ++c2) {
      float dot = 0.f;
#pragma unroll 16
      for (int d = 0; d < 64; ++d) dot += qs[r][d] * ks[c2][d];
      mx = fmaxf(mx, dot * 0.125f);
    }
    // pass 2: exp-sum + weighted accumulation of V
    float acc[64];
#pragma unroll
    for (int d = 0; d < 64; ++d) acc[d] = 0.f;
    float sum = 0.f;
    for (int c2 = 0; c2 < 80; ++c2) {
      float dot = 0.f;
#pragma unroll 16
      for (int d = 0; d < 64; ++d) dot += qs[r][d] * ks[c2][d];
      float e = __expf(dot * 0.125f - mx);
      sum += e;
#pragma unroll
      for (int d = 0; d < 64; ++d) acc[d] += e * vs[c2][d];
    }
    float inv = 1.f / sum;
    int t = r >> 4, sp = r & 15;
    int n = (b * T__ + t) * 16 + sp;
    for (int d = 0; d < 64; ++d)
      y[(size_t)(h * HD_ + d) * NPOS + n] = acc[d] * inv;
  }
}

// ---------------------------------------------------------------------------
// Kernel 6: bilinear upsample 4x4 -> 64x64 (half-pixel) + residual.
// One block per (bt,c) plane; 16 z values cached in LDS. Each thread handles
// 4 consecutive pixels per iteration -> b128 load/store, fully coalesced.
// ---------------------------------------------------------------------------
__global__ __launch_bounds__(256) void upsample_add_kernel(const float* __restrict__ x,
                                                           const float* __restrict__ z, // [256][640]
                                                           float* __restrict__ out) {
  int plane = blockIdx.x;            // bt*256 + c
  int bt = plane >> 8, c = plane & 255;
  __shared__ float tile[16];
  if (threadIdx.x < 16) tile[threadIdx.x] = z[c * NPOS + bt * 16 + threadIdx.x];
  __syncthreads();
  const float* xin = x + (size_t)plane * 4096;
  float* o = out + (size_t)plane * 4096;
#pragma unroll
  for (int i = 0; i < 4; ++i) {
    int base = (i << 10) + (threadIdx.x << 2);   // 4 consecutive pixels
    int yy  = base >> 6;
    int xx0 = base & 63;
    // row interpolation (same y for all 4 pixels)
    float fy  = (yy + 0.5f) * 0.0625f - 0.5f;
    float y0f = floorf(fy);
    float wy  = fy - y0f;
    int y0 = max(0, min(3, (int)y0f));
    int y1 = max(0, min(3, (int)y0f + 1));
    float4 xv = *(const float4*)(xin + base);
    float res[4];
#pragma unroll
    for (int q = 0; q < 4; ++q) {
      float fx  = (xx0 + q + 0.5f) * 0.0625f - 0.5f;
      float x0f = floorf(fx);
      float wx  = fx - x0f;
      int x0 = max(0, min(3, (int)x0f));
      int x1 = max(0, min(3, (int)x0f + 1));
      float v00 = tile[y0 * 4 + x0], v01 = tile[y0 * 4 + x1];
      float v10 = tile[y1 * 4 + x0], v11 = tile[y1 * 4 + x1];
      float v0 = v00 + (v01 - v00) * wx;
      float v1 = v10 + (v11 - v10) * wx;
      res[q] = v0 + (v1 - v0) * wy;
    }
    float4 ov;
    ov.x = xv.x + 0.1f * res[0];
    ov.y = xv.y + 0.1f * res[1];
    ov.z = xv.z + 0.1f * res[2];
    ov.w = xv.w + 0.1f * res[3];
    *(float4*)(o + base) = ov;
  }
}

// ---------------------------------------------------------------------------
extern "C" void kernel_launch(void* const* d_in, const int* in_sizes, int n_in,
                              void* d_out, int out_size, void* d_ws, size_t ws_size,
                              hipStream_t stream) {
  const float* x      = (const float*)d_in[0];
  const float* norm_w = (const float*)d_in[1];
  const float* norm_b = (const float*)d_in[2];
  const float* qkv_w  = (const float*)d_in[3];   // [768][256]
  const float* qkv_b  = (const float*)d_in[4];   // [768]
  const float* proj_w = (const float*)d_in[5];   // [256][256]
  const float* proj_b = (const float*)d_in[6];   // [256]
  const float* pos    = (const float*)d_in[7];   // [T*C] = 1280

  float* ws   = (float*)d_ws;
  float* xp   = ws;                  // [256][640]  pooled
  float* xn   = ws + 163840;         // [256][640]  layernormed
  float* qkv  = ws + 327680;         // [768][640]  QKV output
  float* ybuf = xp;                  // alias: attention output [256][640]
  float* zbuf = xn;                  // alias: proj output      [256][640]
  float* outp = (float*)d_out;

  pool_kernel        <<<BT * C_,        256, 0, stream>>>(x, xp);
  ln_kernel          <<<NPOS,           256, 0, stream>>>(xp, norm_w, norm_b, xn);
  gemm_f32_wmma      <<<(768/16)*(NPOS/16), 32, 0, stream>>>(qkv_w, xn, qkv_b, qkv, 768, NPOS, 256);
  attn_kernel        <<<8 * NH_,        128, 0, stream>>>(qkv, pos, ybuf);
  gemm_f32_wmma      <<<(256/16)*(NPOS/16), 32, 0, stream>>>(proj_w, ybuf, proj_b, zbuf, 256, NPOS, 256);
  upsample_add_kernel<<<BT * C_,        256, 0, stream>>>(x, zbuf, outp);
}